// Extract_21689584845416
// MI455X (gfx1250) — compile-verified
//
#include <hip/hip_runtime.h>
#include <math.h>

// Problem constants (from the reference)
#define CDIM   3
#define IMG_H  2048
#define IMG_W  2048
#define NBOX   256
#define ROIH   32

// Vector type matching the async-LDS builtin's expected parameter type
typedef int v4i_b128 __attribute__((vector_size(4 * sizeof(int))));
typedef __attribute__((address_space(1))) v4i_b128* gptr_b128;
typedef __attribute__((address_space(3))) v4i_b128* lptr_b128;

// Per-box precomputed record (48 bytes, 16B-aligned for b128 loads)
struct __align__(16) BoxRec {
    float x0, y0, x1, y1;
    float x2, y2, x3, y3;
    float widthf;   // float(widths[n])  (truncated int, as float)
    float wm1;      // max(widthf - 1, 1)
    float pad0, pad1;
};

// ---------------------------------------------------------------------------
// Kernel 1: per-box prep. 256 boxes, one wave32-friendly block.
// ---------------------------------------------------------------------------
__global__ void prep_boxes_kernel(const float* __restrict__ boxes,  // [256][4][2]
                                  const float* __restrict__ scale,  // [1]
                                  BoxRec*      __restrict__ recs,   // d_ws
                                  float*       __restrict__ widths_out) // tail of d_out
{
    int n = threadIdx.x;
    if (n >= NBOX) return;

    float s = scale[0];
    const float* b = boxes + (size_t)n * 8;

    // boxes / scale (elementwise IEEE division, as in the reference)
    float x0 = b[0] / s, y0 = b[1] / s;
    float x1 = b[2] / s, y1 = b[3] / s;
    float x2 = b[4] / s, y2 = b[5] / s;
    float x3 = b[6] / s, y3 = b[7] / s;

    // h_dists = ||p2 - p1||, w_dists = ||p1 - p0||
    float hdx = x2 - x1, hdy = y2 - y1;
    float wdx = x1 - x0, wdy = y1 - y0;
    float hd = sqrtf(hdx * hdx + hdy * hdy);
    float wd = sqrtf(wdx * wdx + wdy * wdy);

    // widths = int32(32 * w / max(h, 1e-6))   (truncation, values > 0)
    float wf = (float)ROIH * wd / fmaxf(hd, 1e-6f);
    int   w  = (int)wf;

    BoxRec r;
    r.x0 = x0; r.y0 = y0; r.x1 = x1; r.y1 = y1;
    r.x2 = x2; r.y2 = y2; r.x3 = x3; r.y3 = y3;
    r.widthf = (float)w;
    r.wm1    = fmaxf((float)w - 1.0f, 1.0f);
    r.pad0 = 0.0f; r.pad1 = 0.0f;
    recs[n] = r;

    // widths_out = widths//8*8 + 8  (int values, stored as f32 in d_out tail)
    widths_out[n] = (float)((w / 8) * 8 + 8);
}

// ---------------------------------------------------------------------------
// Kernel 2: main gather. One thread per (box n, row r, col c); writes all
// 3 channels. Output layout: out[n][ch][r][c] (N,C,32,max_w).
//
// CDNA5 specifics:
//  - BoxRec staged to LDS via GLOBAL_LOAD_ASYNC_TO_LDS_B128 (+ s_wait_asynccnt)
//  - global_prefetch_b8 ahead along the sampling row
//  - non-temporal (TH=NT) output stores so the 60MB output stream doesn't
//    evict the 48MB L2-resident image between graph replays
//  - fp contract(off): keep mul/add unfused so rintf() round boundaries
//    match the reference's unfused HLO exactly
// ---------------------------------------------------------------------------
#define HAVE_ASYNC_LDS                                                        \
    (__has_builtin(__builtin_amdgcn_global_load_async_to_lds_b128) &&         \
     __has_builtin(__builtin_amdgcn_s_wait_asynccnt))

__global__ void extract_rows_kernel(const float*  __restrict__ img,  // [3][2048][2048]
                                    const BoxRec* __restrict__ recs,
                                    float*        __restrict__ out,
                                    int max_w)
{
#pragma clang fp contract(off)
    const int c = blockIdx.x * blockDim.x + threadIdx.x;
    const int r = blockIdx.y;   // 0..31
    const int n = blockIdx.z;   // 0..255

#if HAVE_ASYNC_LDS
    // Stage the 48-byte per-box record into LDS once per block:
    // 3 lanes x async b128 copy, wave waits ASYNCcnt, then block barrier.
    __shared__ BoxRec srec;
    {
        char* gsrc = (char*)(recs + n);          // C-cast drops const
        char* ldst = (char*)&srec;
        if (threadIdx.x < 3) {
            // generic -> AS1 / AS3 addrspacecasts via explicit C-style casts
            gptr_b128 gv = (gptr_b128)(v4i_b128*)(gsrc + threadIdx.x * 16);
            lptr_b128 lv = (lptr_b128)(v4i_b128*)(ldst + threadIdx.x * 16);
            __builtin_amdgcn_global_load_async_to_lds_b128(gv, lv, 0, 0);
        }
        __builtin_amdgcn_s_wait_asynccnt(0);   // no-op for non-issuing waves
        __syncthreads();                        // before any early return!
    }
    const BoxRec rec = srec;
#else
    const BoxRec rec = recs[n];
#endif

    if (c >= max_w) return;

    // y = r / 31, x = c / max(width-1, 1)   (true IEEE divisions, as reference)
    const float y = (float)r / (float)(ROIH - 1);
    const float x = (float)c / rec.wm1;
    const float omx = 1.0f - x;
    const float omy = 1.0f - y;

    // bilinear interpolation of the 4 corners (unfused, reference order)
    const float res_x = omx * omy * rec.x0 + x * omy * rec.x1
                      + x   * y   * rec.x2 + omx * y * rec.x3;
    const float res_y = omx * omy * rec.y0 + x * omy * rec.y1
                      + x   * y   * rec.y2 + omx * y * rec.y3;

    const bool mask = ((float)c) < rec.widthf;

    // grid = where(mask, res/(dim-1)*2-1, 2.0); then back-transform + round
    const float gc = mask ? (res_x / (float)(IMG_W - 1) * 2.0f - 1.0f) : 2.0f;
    const float gr = mask ? (res_y / (float)(IMG_H - 1) * 2.0f - 1.0f) : 2.0f;

    // rintf == round-half-to-even (RTE), matching jnp.round
    const float ix = rintf((gc + 1.0f) * (float)(IMG_W - 1) * 0.5f);
    const float iy = rintf((gr + 1.0f) * (float)(IMG_H - 1) * 0.5f);

    const bool inb = (ix >= 0.0f) & (ix <= (float)(IMG_W - 1)) &
                     (iy >= 0.0f) & (iy <= (float)(IMG_H - 1));

    const int ixc = (int)fminf(fmaxf(ix, 0.0f), (float)(IMG_W - 1));
    const int iyc = (int)fminf(fmaxf(iy, 0.0f), (float)(IMG_H - 1));

    const size_t plane = (size_t)IMG_H * (size_t)IMG_W;
    const size_t pix   = (size_t)iyc * IMG_W + (size_t)ixc;

    float v0 = 0.0f, v1 = 0.0f, v2 = 0.0f;
    if (inb) {
        // Speculative prefetch ahead along the sampling row (x advances ~1 px
        // per column) — lowers to global_prefetch_b8 on gfx1250.
        __builtin_prefetch(img + pix + 64, 0, 1);
        v0 = img[pix];
        v1 = img[plane + pix];
        v2 = img[2 * plane + pix];
    }

    // out[n][ch][r][c]
    const size_t chstride = (size_t)ROIH * (size_t)max_w;
    const size_t ob = (((size_t)n * CDIM) * ROIH + (size_t)r) * (size_t)max_w
                      + (size_t)c;
    __builtin_nontemporal_store(v0, out + ob);
    __builtin_nontemporal_store(v1, out + ob + chstride);
    __builtin_nontemporal_store(v2, out + ob + 2 * chstride);
}

// ---------------------------------------------------------------------------
// Launch. Inputs (setup_inputs order): img f32[3*2048*2048], boxes f32[256*4*2],
// scale f32[1]. Output: out f32[256*3*32*max_w] ++ widths_out[256].
// max_w is recovered from out_size (trace-time constant of the reference).
// ---------------------------------------------------------------------------
extern "C" void kernel_launch(void* const* d_in, const int* in_sizes, int n_in,
                              void* d_out, int out_size, void* d_ws, size_t ws_size,
                              hipStream_t stream) {
    const float* img   = (const float*)d_in[0];
    const float* boxes = (const float*)d_in[1];
    const float* scale = (const float*)d_in[2];
    float* out = (float*)d_out;

    const int max_w = (out_size - NBOX) / (NBOX * CDIM * ROIH);

    BoxRec* recs = (BoxRec*)d_ws;                       // 256 * 48 B = 12 KB scratch
    float* widths_out = out + (size_t)NBOX * CDIM * ROIH * (size_t)max_w;

    prep_boxes_kernel<<<1, 256, 0, stream>>>(boxes, scale, recs, widths_out);

    dim3 block(256, 1, 1);                              // 8 wave32 waves
    dim3 grid((max_w + 255) / 256, ROIH, NBOX);
    extract_rows_kernel<<<grid, block, 0, stream>>>(img, recs, out, max_w);
}